// AttentionGNN_6133213298989
// MI455X (gfx1250) — compile-verified
//
#include <hip/hip_runtime.h>
#include <hip/hip_bf16.h>

// Problem constants (match reference)
#define NN   20000
#define EE   640000
#define ENN  660000      // E + N (with self loops)
#define DHH  128
#define DEE  32

typedef __attribute__((ext_vector_type(2))) float v2f;
typedef __attribute__((ext_vector_type(8))) float v8f;

// ---------------------------------------------------------------------------
// Small utility kernels
// ---------------------------------------------------------------------------
__global__ void zero_i32(int* __restrict__ p, int n) {
  int t = blockIdx.x * blockDim.x + threadIdx.x;
  if (t < n) p[t] = 0;
}

__global__ void count_deg(const int* __restrict__ dst, int* __restrict__ deg, int e) {
  int t = blockIdx.x * blockDim.x + threadIdx.x;
  if (t < e) atomicAdd(&deg[dst[t]], 1);
}

// Single-workgroup scan: offsets[n+1] = sum_{m<=n}(deg[m]+1), offsets[0]=0
__global__ __launch_bounds__(1024) void scan_offsets(const int* __restrict__ deg,
                                                     int* __restrict__ offsets, int n) {
  __shared__ int buf[1024];
  __shared__ int carry;
  const int tid = threadIdx.x;
  if (tid == 0) { carry = 0; offsets[0] = 0; }
  __syncthreads();
  for (int base = 0; base < n; base += 1024) {
    int v = (base + tid < n) ? (deg[base + tid] + 1) : 0;
    buf[tid] = v;
    __syncthreads();
    for (int o = 1; o < 1024; o <<= 1) {
      int t = (tid >= o) ? buf[tid - o] : 0;
      __syncthreads();
      buf[tid] += t;
      __syncthreads();
    }
    if (base + tid < n) offsets[base + tid + 1] = carry + buf[tid];
    __syncthreads();
    if (tid == 0) carry += buf[1023];
    __syncthreads();
  }
}

__global__ void fill_csr(const int* __restrict__ src, const int* __restrict__ dst,
                         const int* __restrict__ offsets, int* __restrict__ cursor,
                         int* __restrict__ csr_src, int* __restrict__ csr_eid, int e) {
  int t = blockIdx.x * blockDim.x + threadIdx.x;
  if (t < e) {
    int d = dst[t];
    int pos = offsets[d] + atomicAdd(&cursor[d], 1);
    csr_src[pos] = src[t];
    csr_eid[pos] = t;
  }
}

__global__ void fill_self(const int* __restrict__ offsets,
                          int* __restrict__ csr_src, int* __restrict__ csr_eid, int n) {
  int t = blockIdx.x * blockDim.x + threadIdx.x;
  if (t < n) {
    int pos = offsets[t + 1] - 1;     // self loop occupies the last slot per node
    csr_src[pos] = t;
    csr_eid[pos] = EE + t;
  }
}

// mean edge_attr per dst node; lane == channel (DE==32 == wave32)
__global__ __launch_bounds__(256) void mean_edge_attr(const float* __restrict__ edge_attr,
                                                      const int* __restrict__ offsets,
                                                      const int* __restrict__ csr_eid,
                                                      float* __restrict__ mean_ea, int n) {
  int node = blockIdx.x * 8 + (threadIdx.x >> 5);
  int lane = threadIdx.x & 31;
  if (node >= n) return;
  int start = offsets[node];
  int endr  = offsets[node + 1] - 1;  // exclude self-loop slot
  float s = 0.f;
  for (int pos = start; pos < endr; ++pos) {
    int eid = csr_eid[pos];           // real edge ids < E
    s += edge_attr[(size_t)eid * DEE + lane];
  }
  int deg = endr - start;
  mean_ea[(size_t)node * DEE + lane] = s / (float)(deg > 0 ? deg : 1);
}

// we_ae[l][k] = sum_j We[l][k][j] * att_edge[l][j]   (3x32 outputs)
__global__ void compute_we_ae(const float* __restrict__ We, const float* __restrict__ ae,
                              float* __restrict__ out96) {
  int t = threadIdx.x;
  if (t < 96) {
    int l = t >> 5, k = t & 31;
    const float* wrow = We + ((size_t)l * DEE + k) * DHH;
    const float* a = ae + (size_t)l * DHH;
    float s = 0.f;
    for (int j = 0; j < DHH; ++j) s += wrow[j] * a[j];
    out96[t] = s;
  }
}

// alpha_e[l][e'] = dot(ea[e'], we_ae[l]) for all 3 layers in one pass over edge_attr
__global__ __launch_bounds__(256) void alpha_edge(const float* __restrict__ edge_attr,
                                                  const float* __restrict__ mean_ea,
                                                  const float* __restrict__ we_ae,
                                                  float* __restrict__ alphaE, int total) {
  __shared__ float w[96];
  if (threadIdx.x < 96) w[threadIdx.x] = we_ae[threadIdx.x];
  __syncthreads();
  for (int e = blockIdx.x * blockDim.x + threadIdx.x; e < total;
       e += gridDim.x * blockDim.x) {
    const float* row = (e < EE) ? (edge_attr + (size_t)e * DEE)
                                : (mean_ea + (size_t)(e - EE) * DEE);
    float v[32];
#pragma unroll
    for (int i = 0; i < 8; ++i) {
      float4 t = *(const float4*)(row + i * 4);
      v[i * 4] = t.x; v[i * 4 + 1] = t.y; v[i * 4 + 2] = t.z; v[i * 4 + 3] = t.w;
    }
#pragma unroll
    for (int l = 0; l < 3; ++l) {
      float d = 0.f;
#pragma unroll
      for (int k = 0; k < 32; ++k) d += v[k] * w[l * 32 + k];
      alphaE[(size_t)l * total + e] = d;
    }
  }
}

__global__ void build_h0(const float* __restrict__ x, const float* __restrict__ cx,
                         float* __restrict__ h0, int n) {
  int t = blockIdx.x * blockDim.x + threadIdx.x;
  if (t < n * 64) {
    int node = t >> 6, c = t & 63;
    h0[(size_t)node * DHH + c] = x[t];
    h0[(size_t)node * DHH + 64 + c] = cx[t];
  }
}

// ---------------------------------------------------------------------------
// WMMA GEMM: C[M,128] = A[M,128] @ B[128,128] (+bias). fp32 via 16x16x4 WMMA.
// One wave -> 16 rows x 128 cols (8 accumulators).
// B staged in LDS as interleaved K-pairs: sB[k2*288 + col*2 + (k&1)] so each
// fragment is one aligned ds_load_b64; stride 288 words => the two half-wave
// lane groups hit disjoint 32-bank sets (conflict-free b64 reads).
// ---------------------------------------------------------------------------
__global__ __launch_bounds__(256) void gemm128_wmma(const float* __restrict__ A,
                                                    const float* __restrict__ B,
                                                    const float* __restrict__ bias,
                                                    float* __restrict__ C, int M) {
  __shared__ float sB[32 * 288];               // 36,864 B
  const int tid  = threadIdx.x;
  const int lane = tid & 31;
  const int wave = tid >> 5;
  const int lr   = lane & 15;
  const int hi   = lane >> 4;
  const int mrow = blockIdx.x * 128 + wave * 16;
  const int r    = min(mrow + lr, M - 1);      // clamp: rows >= M never stored

  v8f acc[8];
#pragma unroll
  for (int n = 0; n < 8; ++n) acc[n] = (v8f){0.f, 0.f, 0.f, 0.f, 0.f, 0.f, 0.f, 0.f};

  for (int kh = 0; kh < 2; ++kh) {
    __syncthreads();
    // stage 64x128 chunk of B (rows kh*64 .. kh*64+63) as interleaved k-pairs
#pragma unroll
    for (int i = 0; i < 16; ++i) {
      int p = tid + i * 256;                   // pair index, 4096 total
      int k2 = p >> 7, col = p & 127;
      const float g0 = B[(size_t)(kh * 64 + k2 * 2) * DHH + col];
      const float g1 = B[(size_t)(kh * 64 + k2 * 2 + 1) * DHH + col];
      *(v2f*)(&sB[k2 * 288 + col * 2]) = (v2f){g0, g1};
    }
    __syncthreads();
    for (int ks = 0; ks < 16; ++ks) {
      // A frag: lanes 0-15 -> K = kb,kb+1 ; lanes 16-31 -> K = kb+2,kb+3
      v2f a = *(const v2f*)(A + (size_t)r * DHH + kh * 64 + ks * 4 + 2 * hi);
      const int krow = ks * 2 + hi;            // pair-row in LDS
#pragma unroll
      for (int n = 0; n < 8; ++n) {
        v2f b = *(const v2f*)(&sB[krow * 288 + (n * 16 + lr) * 2]);
        acc[n] = __builtin_amdgcn_wmma_f32_16x16x4_f32(
            false, a, false, b, (short)0, acc[n], false, false);
      }
    }
  }
  // C/D layout: acc[n][g] -> row = mrow + g + 8*hi, col = n*16 + lr
#pragma unroll
  for (int n = 0; n < 8; ++n) {
    const int col = n * 16 + lr;
    const float bv = bias ? bias[col] : 0.f;
#pragma unroll
    for (int g = 0; g < 8; ++g) {
      const int row = mrow + g + 8 * hi;
      if (row < M) C[(size_t)row * DHH + col] = acc[n][g] + bv;
    }
  }
}

// ---------------------------------------------------------------------------
// s_src[n]=dot(h'[n],a_s), s_dst[n]=dot(h'[n],a_d). One wave per node.
// ---------------------------------------------------------------------------
__global__ __launch_bounds__(256) void node_dots(const float* __restrict__ h,
                                                 const float* __restrict__ a_s,
                                                 const float* __restrict__ a_d,
                                                 float* __restrict__ ssrc,
                                                 float* __restrict__ sdst, int n) {
  int node = blockIdx.x * 8 + (threadIdx.x >> 5);
  int lane = threadIdx.x & 31;
  if (node >= n) return;
  float4 hv = *(const float4*)(h + (size_t)node * DHH + lane * 4);
  float4 as = *(const float4*)(a_s + lane * 4);
  float4 ad = *(const float4*)(a_d + lane * 4);
  float ds = hv.x * as.x + hv.y * as.y + hv.z * as.z + hv.w * as.w;
  float dd = hv.x * ad.x + hv.y * ad.y + hv.z * ad.z + hv.w * ad.w;
  for (int o = 16; o; o >>= 1) {
    ds += __shfl_xor(ds, o, 32);
    dd += __shfl_xor(dd, o, 32);
  }
  if (lane == 0) { ssrc[node] = ds; sdst[node] = dd; }
}

// ---------------------------------------------------------------------------
// Softmax-attention aggregation, CSR by destination. One wave per dst node.
// Pass 1: running max. Pass 2 (fused): per 32-edge chunk each lane computes
// one edge's unnormalized weight w and src; __shfl broadcasts drive the
// channel-parallel accumulation of sum(w*h'[src]) while denom accumulates.
// Final: out = acc/denom + bias (optional ReLU).
// ---------------------------------------------------------------------------
__global__ __launch_bounds__(256) void gat_aggregate(const float* __restrict__ hprime,
                                                     const float* __restrict__ ssrc,
                                                     const float* __restrict__ sdst,
                                                     const float* __restrict__ alphaE,
                                                     const int* __restrict__ csr_src,
                                                     const int* __restrict__ csr_eid,
                                                     const int* __restrict__ offsets,
                                                     const float* __restrict__ bias,
                                                     float* __restrict__ out, int relu) {
  int node = blockIdx.x * 8 + (threadIdx.x >> 5);
  int lane = threadIdx.x & 31;
  if (node >= NN) return;
  const int start = offsets[node];
  const int end   = offsets[node + 1];
  const float sd  = sdst[node];

  // pass 1: max score (lanes stride over edges)
  float m = -1e30f;
  for (int pos = start + lane; pos < end; pos += 32) {
    float sc = ssrc[csr_src[pos]] + sd + alphaE[csr_eid[pos]];
    sc = sc > 0.f ? sc : 0.2f * sc;
    m = fmaxf(m, sc);
  }
  for (int o = 16; o; o >>= 1) m = fmaxf(m, __shfl_xor(m, o, 32));

  // pass 2: fused weight + accumulation (unnormalized), chunked by 32 edges
  const int cb = lane * 4;
  float denom = 0.f;
  float4 acc = make_float4(0.f, 0.f, 0.f, 0.f);
  for (int chunk = start; chunk < end; chunk += 32) {
    const int pos = chunk + lane;
    float w = 0.f;
    int s = 0;
    if (pos < end) {
      s = csr_src[pos];
      float sc = ssrc[s] + sd + alphaE[csr_eid[pos]];
      sc = sc > 0.f ? sc : 0.2f * sc;
      w = __expf(sc - m);
    }
    denom += w;
    const int cnt = min(32, end - chunk);
    for (int j = 0; j < cnt; ++j) {
      const float wj = __shfl(w, j, 32);
      const int   sj = __shfl(s, j, 32);
      float4 hv = *(const float4*)(hprime + (size_t)sj * DHH + cb);
      acc.x += wj * hv.x; acc.y += wj * hv.y;
      acc.z += wj * hv.z; acc.w += wj * hv.w;
    }
  }
  for (int o = 16; o; o >>= 1) denom += __shfl_xor(denom, o, 32);
  const float inv = 1.0f / denom;

  float4 bv = *(const float4*)(bias + cb);
  float4 o4;
  o4.x = acc.x * inv + bv.x; o4.y = acc.y * inv + bv.y;
  o4.z = acc.z * inv + bv.z; o4.w = acc.w * inv + bv.w;
  if (relu) {
    o4.x = fmaxf(o4.x, 0.f); o4.y = fmaxf(o4.y, 0.f);
    o4.z = fmaxf(o4.z, 0.f); o4.w = fmaxf(o4.w, 0.f);
  }
  *(float4*)(out + (size_t)node * DHH + cb) = o4;
}

// ---------------------------------------------------------------------------
extern "C" void kernel_launch(void* const* d_in, const int* in_sizes, int n_in,
                              void* d_out, int out_size, void* d_ws, size_t ws_size,
                              hipStream_t stream) {
  const float* x        = (const float*)d_in[0];
  const float* cx       = (const float*)d_in[1];
  const int*   eidx     = (const int*)d_in[2];
  const float* eattr    = (const float*)d_in[3];
  const float* Ws       = (const float*)d_in[4];
  const float* att_src  = (const float*)d_in[5];
  const float* att_dst  = (const float*)d_in[6];
  const float* We       = (const float*)d_in[7];
  const float* att_edge = (const float*)d_in[8];
  const float* bias     = (const float*)d_in[9];
  const float* lin_W    = (const float*)d_in[10];
  const float* lin_b    = (const float*)d_in[11];
  (void)in_sizes; (void)n_in; (void)out_size; (void)ws_size;

  const int* src0 = eidx;
  const int* dst0 = eidx + EE;

  char* ws = (char*)d_ws;
  size_t off = 0;
  auto alloc = [&](size_t bytes) -> void* {
    void* p = ws + off;
    off = (off + bytes + 255) & ~(size_t)255;
    return p;
  };
  int*   deg     = (int*)alloc((size_t)NN * 4);
  int*   cursor  = (int*)alloc((size_t)NN * 4);
  int*   offsets = (int*)alloc((size_t)(NN + 1) * 4);
  int*   csr_src = (int*)alloc((size_t)ENN * 4);
  int*   csr_eid = (int*)alloc((size_t)ENN * 4);
  float* mean_ea = (float*)alloc((size_t)NN * DEE * 4);
  float* we_ae   = (float*)alloc(96 * 4);
  float* alphaE  = (float*)alloc((size_t)3 * ENN * 4);
  float* ssrc    = (float*)alloc((size_t)NN * 4);
  float* sdst    = (float*)alloc((size_t)NN * 4);
  float* bufA    = (float*)alloc((size_t)NN * DHH * 4);
  float* bufB    = (float*)alloc((size_t)NN * DHH * 4);

  // --- graph preprocessing (deterministic every call) ---
  zero_i32<<<(NN + 255) / 256, 256, 0, stream>>>(deg, NN);
  zero_i32<<<(NN + 255) / 256, 256, 0, stream>>>(cursor, NN);
  count_deg<<<(EE + 255) / 256, 256, 0, stream>>>(dst0, deg, EE);
  scan_offsets<<<1, 1024, 0, stream>>>(deg, offsets, NN);
  fill_csr<<<(EE + 255) / 256, 256, 0, stream>>>(src0, dst0, offsets, cursor,
                                                 csr_src, csr_eid, EE);
  fill_self<<<(NN + 255) / 256, 256, 0, stream>>>(offsets, csr_src, csr_eid, NN);
  mean_edge_attr<<<(NN + 7) / 8, 256, 0, stream>>>(eattr, offsets, csr_eid, mean_ea, NN);
  compute_we_ae<<<1, 128, 0, stream>>>(We, att_edge, we_ae);
  alpha_edge<<<(ENN + 255) / 256, 256, 0, stream>>>(eattr, mean_ea, we_ae, alphaE, ENN);
  build_h0<<<(NN * 64 + 255) / 256, 256, 0, stream>>>(x, cx, bufA, NN);

  // --- 3 GAT layers ---
  for (int l = 0; l < 3; ++l) {
    gemm128_wmma<<<(NN + 127) / 128, 256, 0, stream>>>(
        bufA, Ws + (size_t)l * DHH * DHH, nullptr, bufB, NN);
    node_dots<<<(NN + 7) / 8, 256, 0, stream>>>(
        bufB, att_src + (size_t)l * DHH, att_dst + (size_t)l * DHH, ssrc, sdst, NN);
    gat_aggregate<<<(NN + 7) / 8, 256, 0, stream>>>(
        bufB, ssrc, sdst, alphaE + (size_t)l * ENN, csr_src, csr_eid, offsets,
        bias + (size_t)l * DHH, bufA, (l < 2) ? 1 : 0);
  }

  // --- final linear: out = h @ lin_W + lin_b ---
  gemm128_wmma<<<(NN + 127) / 128, 256, 0, stream>>>(
      bufA, lin_W, lin_b, (float*)d_out, NN);
}